// MTRNN_14413910245613
// MI455X (gfx1250) — compile-verified
//
#include <hip/hip_runtime.h>
#include <math.h>

// MTRNN on gfx1250 (MI455X).
// - Serial recurrence => parallelize over batch only: 8 workgroups x 16-row tiles,
//   8 wave32s each; waves split the 1024 output columns (8 x 16-col N-tiles).
// - f32 WMMA (V_WMMA_F32_16X16X4_F32): numerically matches the f32 reference.
// - Weights repacked once per launch into d_ws as Wpack[k/4][j][k%4] so each
//   B fragment is ONE aligned global_load_b64 (wave reads 256B contiguous),
//   with pure-immediate kc offsets. Combined 4MB matrix stays L2-resident.
// - Activations (tanh of state) shared via LDS in the same K-packed layout:
//   ds_load_b64 A-fragments hit all 64 banks exactly once, addresses linear in kc.
// - Hidden state lives in registers in the WMMA C-matrix lane layout.

typedef float v2f __attribute__((ext_vector_type(2)));
typedef float v8f __attribute__((ext_vector_type(8)));

#define DD   512
#define OO   64
#define TT   1024
#define BB   128
#define NN   1024   // combined state width (2*DD)
#define NT   8      // N-tiles (of 16 cols) per wave
#define NWAVES 8

// ---------------------------------------------------------------------------
// Pre-pass: build K-packed combined weight matrix in workspace.
//   Wpack[kc][j][kl] = W[kc*4+kl][j],  W = [[w_ss, w_sf],[w_fs, w_ff]]
//   WOpack[kc][n][kl] = w_o[n][kc*4+kl]        (the output projection, w_o^T)
// Writes are fully coalesced (linear idx); reads strided but this is 4MB once.
// ---------------------------------------------------------------------------
__global__ void __launch_bounds__(256)
mtrnn_prepack(const float* __restrict__ w_ss, const float* __restrict__ w_ff,
              const float* __restrict__ w_sf, const float* __restrict__ w_fs,
              const float* __restrict__ w_o,
              float* __restrict__ wpack, float* __restrict__ wopack)
{
  const int stride = gridDim.x * blockDim.x;
  const int tid0   = blockIdx.x * blockDim.x + threadIdx.x;

  for (int idx = tid0; idx < NN * NN; idx += stride) {
    const int kc = idx >> 12;          // 0..255
    const int j  = (idx >> 2) & (NN - 1);
    const int kl = idx & 3;
    const int k  = kc * 4 + kl;        // 0..1023
    const float* q;
    int kk = k, jj = j;
    if (k < DD) { if (j < DD) q = w_ss; else { q = w_sf; jj = j - DD; } }
    else { kk = k - DD; if (j < DD) q = w_fs; else { q = w_ff; jj = j - DD; } }
    wpack[idx] = q[kk * DD + jj];
  }
  for (int idx = tid0; idx < DD * OO; idx += stride) {
    const int kc = idx >> 8;           // 0..127
    const int n  = (idx >> 2) & (OO - 1);
    const int kl = idx & 3;
    wopack[idx] = w_o[n * DD + kc * 4 + kl];
  }
}

// ---------------------------------------------------------------------------
// Main recurrent kernel. One workgroup per 16-row batch tile.
// ---------------------------------------------------------------------------
__global__ void __launch_bounds__(256, 1)
mtrnn_kernel(const float* __restrict__ xin,
             const float* __restrict__ hs_init,
             const float* __restrict__ hf_init,
             const float* __restrict__ b_o,
             const float* __restrict__ b_s,
             const float* __restrict__ b_f,
             const float* __restrict__ wpack,
             const float* __restrict__ wopack,
             float* __restrict__ out)
{
  // Apack[kc][m][kl] : 256 x 16 x 4 f32 = 64KB. Fragment read at (m, 4kc+rs..+1)
  // is dword offset kc*64 + m*4 + rs -> ds_load_b64, 64 banks covered once.
  extern __shared__ float Alds[];

  const int tid    = threadIdx.x;
  const int wave   = tid >> 5;             // 0..7
  const int lane   = tid & 31;
  const int nl     = lane & 15;            // N (or M for A-frag) within tile
  const int khi    = lane >> 4;            // 0/1 -> which K pair
  const int rs     = khi * 2;              // K row-pair base {0,2}
  const int batch0 = blockIdx.x * 16;

  float        bias[NT], decay[NT], invt[NT];
  const float* bptr[NT];                   // per-lane B-frag base in wpack
  int          abase[NT];                  // per-lane A-build LDS dword base
  v8f          st[NT];                     // state, C-matrix lane layout

  for (int i = 0; i < NT; ++i) {
    const int ti = wave + NWAVES * i;      // 0..63
    const int n0 = ti * 16;
    const int c  = n0 + nl;                // combined column 0..1023
    const bool slow = (n0 < DD);
    const int  jl   = slow ? c : (c - DD);
    bias[i]  = slow ? b_s[jl] : b_f[jl];
    invt[i]  = slow ? (1.0f / 20.0f) : (1.0f / 2.0f);
    decay[i] = 1.0f - invt[i];
    bptr[i]  = wpack + (size_t)c * 4 + rs; // + kc*4096 floats per K-chunk
    abase[i] = (c >> 2) * 64 + (c & 3);    // + m*4 per row
    const float* initp = slow ? hs_init : hf_init;
#pragma unroll
    for (int r = 0; r < 8; ++r) {
      const int m = r + 8 * khi;
      st[i][r] = initp[(batch0 + m) * DD + jl];
    }
  }
  const float  bo   = (wave < 4) ? b_o[wave * 16 + nl] : 0.0f;
  const float* wop  = wopack + (size_t)(wave * 16 + nl) * 4 + rs; // + kc*256

  for (int t = 0; t <= TT; ++t) {
    // ---- build A = tanh(state) into K-packed LDS (each wave its columns)
    for (int i = 0; i < NT; ++i) {
      const int ab = abase[i];
#pragma unroll
      for (int r = 0; r < 8; ++r) {
        const int m = r + 8 * khi;
        Alds[ab + m * 4] = tanhf(st[i][r]);
      }
    }
    __syncthreads();

    // ---- output projection for step t-1: tanh(hf) == A columns [512,1024)
    if (t > 0 && wave < 4) {
      v8f acc = {0.f, 0.f, 0.f, 0.f, 0.f, 0.f, 0.f, 0.f};
#pragma unroll 4
      for (int kc = 0; kc < DD / 4; ++kc) {
        v2f a = *(const v2f*)&Alds[(128 + kc) * 64 + nl * 4 + rs];
        v2f b = *(const v2f*)(wop + (size_t)kc * 256);
        acc = __builtin_amdgcn_wmma_f32_16x16x4_f32(
            false, a, false, b, (short)0, acc, false, false);
      }
      const int n0 = wave * 16;
      const float* xt = xin + ((size_t)(t - 1) * BB + batch0) * OO;
      float*       ot = out + ((size_t)(t - 1) * BB + batch0) * OO;
#pragma unroll
      for (int r = 0; r < 8; ++r) {
        const int m = r + 8 * khi;
        ot[m * OO + n0 + nl] = acc[r] + bo - xt[m * OO + n0 + nl];
      }
    }
    if (t == TT) break;

    // ---- main GEMM: C[16x1024] = A[16x1024] @ W[1024x1024], this wave's tiles
    v8f acc[NT];
#pragma unroll
    for (int i = 0; i < NT; ++i) acc[i] = (v8f){0.f,0.f,0.f,0.f,0.f,0.f,0.f,0.f};

#pragma unroll 4
    for (int kc = 0; kc < NN / 4; ++kc) {
      v2f a = *(const v2f*)&Alds[kc * 64 + nl * 4 + rs];
#pragma unroll
      for (int i = 0; i < NT; ++i) {
        v2f b = *(const v2f*)(bptr[i] + (size_t)kc * 4096);
        acc[i] = __builtin_amdgcn_wmma_f32_16x16x4_f32(
            false, a, false, b, (short)0, acc[i], false, false);
      }
    }

    // ---- leaky-integrator blend into register state
#pragma unroll
    for (int i = 0; i < NT; ++i) {
#pragma unroll
      for (int r = 0; r < 8; ++r)
        st[i][r] = decay[i] * st[i][r] + invt[i] * (acc[i][r] + bias[i]);
    }
    __syncthreads();   // all waves done reading A before next step rewrites it
  }
}

extern "C" void kernel_launch(void* const* d_in, const int* in_sizes, int n_in,
                              void* d_out, int out_size, void* d_ws, size_t ws_size,
                              hipStream_t stream) {
  const float* xx      = (const float*)d_in[0];
  const float* hs_init = (const float*)d_in[1];
  const float* hf_init = (const float*)d_in[2];
  const float* w_o     = (const float*)d_in[3];
  const float* b_o     = (const float*)d_in[4];
  const float* w_ss    = (const float*)d_in[5];
  const float* b_s     = (const float*)d_in[6];
  const float* w_ff    = (const float*)d_in[7];
  const float* b_f     = (const float*)d_in[8];
  const float* w_sf    = (const float*)d_in[9];
  const float* w_fs    = (const float*)d_in[10];

  float* wpack  = (float*)d_ws;                       // 1024*1024 f32 = 4MB
  float* wopack = wpack + (size_t)NN * NN;            // 512*64 f32 = 128KB

  mtrnn_prepack<<<512, 256, 0, stream>>>(w_ss, w_ff, w_sf, w_fs, w_o,
                                         wpack, wopack);

  dim3 grid(BB / 16);   // 8 batch tiles
  dim3 block(256);      // 8 wave32s
  size_t lds = 16 * 1024 * sizeof(float);   // 64KB K-packed activation buffer
  mtrnn_kernel<<<grid, block, lds, stream>>>(
      xx, hs_init, hf_init, b_o, b_s, b_f, wpack, wopack, (float*)d_out);
}